// TransformerModel_7859790152091
// MI455X (gfx1250) — compile-verified
//
#include <hip/hip_runtime.h>
#include <math.h>

// ---------------- model constants (match reference) ----------------
#define NL      2
#define NH      8
#define DMODEL  512
#define DHEAD   64
#define DFFN    2048
#define NVOCAB  32000
#define BATCH   8
#define SSRC    512
#define STGT    256

typedef float v2f __attribute__((ext_vector_type(2)));
typedef float v8f __attribute__((ext_vector_type(8)));

// =====================================================================
// WMMA fp32 GEMM:  C = alpha * A @ op(B) + bias + residual, opt. ReLU
//   BT=1 : op(B) = B^T, B stored [N,K] row-major (weights, K-matrix)
//   BT=0 : op(B) = B,   B stored [K,N] row-major (V-matrix)
// Block = 128 threads (4 waves); tile 128(M) x 64(N); K step 16.
// Each wave owns 32 rows (two 16-row A fragments sharing each B fragment)
// -> 8 v_wmma per k-step vs 6 LDS fragment loads.
// B tile stored k-pair-interleaved (lds_b[kp][n][2]) so a B fragment is a
// single aligned ds_load_b64 into the WMMA source register pair.
// blockIdx.z batch slice: zb=z/zInner, zi=z%zInner; ptr += zb*sXo + zi*sXi.
// Requires: M%128==0, N%64==0, K%16==0 (true for every GEMM in this model).
// =====================================================================
template <int BT>
__global__ __launch_bounds__(128) void gemm_wmma_kernel(
    const float* __restrict__ A, int lda,
    const float* __restrict__ Bm, int ldb,
    float* C, int ldc,
    const float* __restrict__ bias,
    const float* resid,
    int K, float alpha, int relu, int zInner,
    long sAo, long sAi, long sBo, long sBi, long sCo, long sCi)
{
    __shared__ __align__(16) float lds_a[128][20];     // 128x16 tile, +4 pad
    __shared__ __align__(16) float lds_b[8][65][2];    // 16x64 tile, k-pair interleaved

    const int tid  = threadIdx.x;
    const int lane = tid & 31;
    const int wid  = tid >> 5;        // wave 0..3 -> rows [wid*32, wid*32+32)
    const int lo   = lane & 15;
    const int hi   = lane >> 4;       // half-wave: K sub-pair / C row half

    const int m0 = blockIdx.y * 128;
    const int n0 = blockIdx.x * 64;
    const int zb = blockIdx.z / zInner;
    const int zi = blockIdx.z % zInner;

    const float* Ap = A  + (size_t)zb * sAo + (size_t)zi * sAi;
    const float* Bp = Bm + (size_t)zb * sBo + (size_t)zi * sBi;
    float*       Cp = C  + (size_t)zb * sCo + (size_t)zi * sCi;
    const float* Rp = resid ? (resid + (size_t)zb * sCo + (size_t)zi * sCi)
                            : (const float*)0;

    v8f acc[2][4] = {};

    // staging thread mapping
    const int wr = tid >> 1;          // B(BT) row within N tile (0..63)
    const int ac = (tid & 1) * 8;     // k offset {0,8}
    const int br = tid >> 3;          // B(K-major) row (0..15)
    const int bc = (tid & 7) * 8;     // B col offset {0..56}

    for (int k0 = 0; k0 < K; k0 += 16) {
        __syncthreads();
        {   // stage A 128x16: one row per thread, 16 floats (4x b128)
            const float* s = Ap + (size_t)(m0 + tid) * lda + k0;
            float4 u0 = *(const float4*)(s);
            float4 u1 = *(const float4*)(s + 4);
            float4 u2 = *(const float4*)(s + 8);
            float4 u3 = *(const float4*)(s + 12);
            *(float4*)&lds_a[tid][0]  = u0;
            *(float4*)&lds_a[tid][4]  = u1;
            *(float4*)&lds_a[tid][8]  = u2;
            *(float4*)&lds_a[tid][12] = u3;
        }
        if (BT) {   // stage B^T: read W[n][k] rows, store k-pair interleaved
            const float* s = Bp + (size_t)(n0 + wr) * ldb + k0 + ac;
            float4 u0 = *(const float4*)(s);
            float4 u1 = *(const float4*)(s + 4);
            const int kp = ac >> 1;
            v2f p;
            p.x = u0.x; p.y = u0.y; *(v2f*)&lds_b[kp + 0][wr][0] = p;
            p.x = u0.z; p.y = u0.w; *(v2f*)&lds_b[kp + 1][wr][0] = p;
            p.x = u1.x; p.y = u1.y; *(v2f*)&lds_b[kp + 2][wr][0] = p;
            p.x = u1.z; p.y = u1.w; *(v2f*)&lds_b[kp + 3][wr][0] = p;
        } else {    // stage B rows, scatter into pair-interleaved layout
            const float* s = Bp + (size_t)(k0 + br) * ldb + n0 + bc;
            float4 u0 = *(const float4*)(s);
            float4 u1 = *(const float4*)(s + 4);
            const int kp = br >> 1, j = br & 1;
            lds_b[kp][bc + 0][j] = u0.x;  lds_b[kp][bc + 1][j] = u0.y;
            lds_b[kp][bc + 2][j] = u0.z;  lds_b[kp][bc + 3][j] = u0.w;
            lds_b[kp][bc + 4][j] = u1.x;  lds_b[kp][bc + 5][j] = u1.y;
            lds_b[kp][bc + 6][j] = u1.z;  lds_b[kp][bc + 7][j] = u1.w;
        }
        __syncthreads();

        #pragma unroll
        for (int kk = 0; kk < 16; kk += 4) {
            const int kb = kk + 2 * hi;       // ISA A/B fragment K layout
            const int kp = kb >> 1;
            const v2f a0 = *(const v2f*)&lds_a[wid * 32 + lo][kb];
            const v2f a1 = *(const v2f*)&lds_a[wid * 32 + 16 + lo][kb];
            #pragma unroll
            for (int t = 0; t < 4; ++t) {
                const v2f b = *(const v2f*)&lds_b[kp][t * 16 + lo][0];
                acc[0][t] = __builtin_amdgcn_wmma_f32_16x16x4_f32(
                    false, a0, false, b, (short)0, acc[0][t], false, false);
                acc[1][t] = __builtin_amdgcn_wmma_f32_16x16x4_f32(
                    false, a1, false, b, (short)0, acc[1][t], false, false);
            }
        }
    }

    // epilogue: C layout = VGPR v, half-wave hi -> row 8*hi+v, col lo
    #pragma unroll
    for (int r = 0; r < 2; ++r)
        #pragma unroll
        for (int t = 0; t < 4; ++t)
            #pragma unroll
            for (int v = 0; v < 8; ++v) {
                const int m = m0 + wid * 32 + r * 16 + hi * 8 + v;
                const int n = n0 + t * 16 + lo;
                float val = alpha * acc[r][t][v];
                if (bias) val += bias[n];
                if (Rp)   val += Rp[(size_t)m * ldc + n];
                if (relu) val = fmaxf(val, 0.f);
                Cp[(size_t)m * ldc + n] = val;
            }
}

// =====================================================================
// Embedding + sinusoidal positional encoding:
//   out[t,d] = emb[tok[t],d]*sqrt(D) + PE(t%S, d)
// =====================================================================
__global__ void embed_kernel(const int* __restrict__ tok,
                             const float* __restrict__ emb,
                             float* __restrict__ out,
                             int T, int S, float scale)
{
    long i = (long)blockIdx.x * blockDim.x + threadIdx.x;
    if (i >= (long)T * DMODEL) return;
    int t = (int)(i / DMODEL), d = (int)(i % DMODEL);
    int pos = t % S;
    int half = d >> 1;
    float freq = expf((float)(2 * half) * (-9.210340371976184f / (float)DMODEL));
    float ang  = (float)pos * freq;
    float pe   = (d & 1) ? cosf(ang) : sinf(ang);
    out[i] = emb[(size_t)tok[t] * DMODEL + d] * scale + pe;
}

// =====================================================================
// LayerNorm (reference-faithful: unbiased var/(D-1), divide by std+eps)
// one 128-thread block per row of D=512
// =====================================================================
__global__ __launch_bounds__(128) void layernorm_kernel(
    const float* __restrict__ x, const float* __restrict__ ga,
    const float* __restrict__ gb, float* __restrict__ out)
{
    __shared__ float red[128];
    const int row = blockIdx.x, tid = threadIdx.x;
    const float* xr = x + (size_t)row * DMODEL;
    float s = 0.f;
    for (int i = tid; i < DMODEL; i += 128) s += xr[i];
    red[tid] = s; __syncthreads();
    for (int o = 64; o > 0; o >>= 1) { if (tid < o) red[tid] += red[tid + o]; __syncthreads(); }
    const float mean = red[0] / (float)DMODEL;
    __syncthreads();
    float v = 0.f;
    for (int i = tid; i < DMODEL; i += 128) { float c = xr[i] - mean; v += c * c; }
    red[tid] = v; __syncthreads();
    for (int o = 64; o > 0; o >>= 1) { if (tid < o) red[tid] += red[tid + o]; __syncthreads(); }
    const float inv = 1.f / (sqrtf(red[0] / (float)(DMODEL - 1)) + 1e-6f);
    __syncthreads();
    float* orow = out + (size_t)row * DMODEL;
    for (int i = tid; i < DMODEL; i += 128)
        orow[i] = ga[i] * (xr[i] - mean) * inv + gb[i];
}

// =====================================================================
// Masked softmax over last dim of scores [B,H,Sq,Sk] (in-place).
// mask (optional): int [B,Sq,Sk]; mask==0 -> score := -1e9 (pre-softmax)
// =====================================================================
__global__ __launch_bounds__(128) void softmax_kernel(
    float* sc, const int* __restrict__ mask, int Sq, int Sk)
{
    __shared__ float red[128];
    const int row = blockIdx.x, tid = threadIdx.x;
    float* r = sc + (size_t)row * Sk;
    const int* mrow = (const int*)0;
    if (mask) {
        int b = row / (NH * Sq);
        int q = row % Sq;
        mrow = mask + ((size_t)b * Sq + q) * Sk;
    }
    float mx = -3.0e38f;
    for (int i = tid; i < Sk; i += 128) {
        float v = r[i];
        if (mrow && mrow[i] == 0) v = -1e9f;
        mx = fmaxf(mx, v);
    }
    red[tid] = mx; __syncthreads();
    for (int o = 64; o > 0; o >>= 1) { if (tid < o) red[tid] = fmaxf(red[tid], red[tid + o]); __syncthreads(); }
    mx = red[0]; __syncthreads();
    float s = 0.f;
    for (int i = tid; i < Sk; i += 128) {
        float v = r[i];
        if (mrow && mrow[i] == 0) v = -1e9f;
        float e = expf(v - mx);
        r[i] = e;
        s += e;
    }
    red[tid] = s; __syncthreads();
    for (int o = 64; o > 0; o >>= 1) { if (tid < o) red[tid] += red[tid + o]; __syncthreads(); }
    const float inv = 1.f / red[0];
    __syncthreads();
    for (int i = tid; i < Sk; i += 128) r[i] *= inv;
}

// =====================================================================
// Row-wise log-softmax over NVOCAB, in place on d_out
// =====================================================================
__global__ __launch_bounds__(256) void logsoftmax_kernel(float* lg)
{
    __shared__ float red[256];
    const int row = blockIdx.x, tid = threadIdx.x;
    float* r = lg + (size_t)row * NVOCAB;
    float mx = -3.0e38f;
    for (int i = tid; i < NVOCAB; i += 256) mx = fmaxf(mx, r[i]);
    red[tid] = mx; __syncthreads();
    for (int o = 128; o > 0; o >>= 1) { if (tid < o) red[tid] = fmaxf(red[tid], red[tid + o]); __syncthreads(); }
    mx = red[0]; __syncthreads();
    float s = 0.f;
    for (int i = tid; i < NVOCAB; i += 256) s += expf(r[i] - mx);
    red[tid] = s; __syncthreads();
    for (int o = 128; o > 0; o >>= 1) { if (tid < o) red[tid] += red[tid + o]; __syncthreads(); }
    const float lse = mx + logf(red[0]);
    __syncthreads();
    for (int i = tid; i < NVOCAB; i += 256) r[i] = r[i] - lse;
}

// ---------------- host-side parameter views ----------------
struct LinP  { const float *w, *b; };
struct LNP   { const float *a, *b; };
struct AttnP { LinP q, k, v, o; };
struct EncL  { AttnP sa; LNP sa_ln; LinP l1, l2; LNP ff_ln; };
struct DecL  { AttnP sa; LNP sa_ln; AttnP ca; LNP ca_ln; LinP l1, l2; LNP ff_ln; };

extern "C" void kernel_launch(void* const* d_in, const int* in_sizes, int n_in,
                              void* d_out, int out_size, void* d_ws, size_t ws_size,
                              hipStream_t stream)
{
    (void)in_sizes; (void)out_size; (void)ws_size;
    if (n_in < 95) return;   // expect 92 param leaves + src + tgt + mask

    // ---- unpack in setup_inputs() insertion order, depth-first ----
    int ix = 0;
    auto F       = [&]() { return (const float*)d_in[ix++]; };
    auto getLin  = [&]() { LinP  l; l.w = F(); l.b = F(); return l; };
    auto getLN   = [&]() { LNP   l; l.a = F(); l.b = F(); return l; };
    auto getAttn = [&]() { AttnP a; a.q = getLin(); a.k = getLin();
                                    a.v = getLin(); a.o = getLin(); return a; };

    const float* src_emb = F();
    const float* tgt_emb = F();
    EncL enc[NL];
    for (int l = 0; l < NL; ++l) {
        enc[l].sa = getAttn(); enc[l].sa_ln = getLN();
        enc[l].l1 = getLin();  enc[l].l2 = getLin(); enc[l].ff_ln = getLN();
    }
    LNP enc_lnp = getLN();
    DecL dec[NL];
    for (int l = 0; l < NL; ++l) {
        dec[l].sa = getAttn(); dec[l].sa_ln = getLN();
        dec[l].ca = getAttn(); dec[l].ca_ln = getLN();
        dec[l].l1 = getLin();  dec[l].l2 = getLin(); dec[l].ff_ln = getLN();
    }
    LNP  dec_lnp = getLN();
    LinP genp    = getLin();
    const int* srcTok  = (const int*)d_in[ix++];
    const int* tgtTok  = (const int*)d_in[ix++];
    const int* decMask = (const int*)d_in[ix++];

    // ---- workspace bump allocator ----
    const size_t Te = (size_t)BATCH * SSRC;   // 4096 encoder tokens
    const size_t Td = (size_t)BATCH * STGT;   // 2048 decoder tokens
    char* wp = (char*)d_ws;
    auto alloc = [&](size_t nf) {
        float* p = (float*)wp;
        wp += ((nf * sizeof(float) + 255) & ~(size_t)255);
        return p;
    };
    float* XE  = alloc(Te * DMODEL);                      // encoder residual stream
    float* XD  = alloc(Td * DMODEL);                      // decoder residual stream
    float* NBF = alloc(Te * DMODEL);                      // layernorm output
    float* Qb  = alloc(Te * DMODEL);
    float* Kb  = alloc(Te * DMODEL);
    float* Vb  = alloc(Te * DMODEL);
    float* ATT = alloc(Te * DMODEL);                      // attention ctx pre-O
    float* MEM = alloc(Te * DMODEL);                      // encoder memory
    float* FFH = alloc(Te * DFFN);                        // FFN hidden
    float* SC  = alloc((size_t)BATCH * NH * SSRC * SSRC); // attention scores

    // ---- launchers ----
    auto gemm = [&](bool bt, const float* A, int lda, const float* Bm, int ldb,
                    float* C, int ldc, int M, int N, int K,
                    const float* bias, const float* resid, float alpha, bool relu,
                    int nz, int zInner,
                    long sAo, long sAi, long sBo, long sBi, long sCo, long sCi) {
        dim3 g(N / 64, M / 128, nz), b(128, 1, 1);
        if (bt)
            gemm_wmma_kernel<1><<<g, b, 0, stream>>>(A, lda, Bm, ldb, C, ldc,
                bias, resid, K, alpha, relu ? 1 : 0, zInner,
                sAo, sAi, sBo, sBi, sCo, sCi);
        else
            gemm_wmma_kernel<0><<<g, b, 0, stream>>>(A, lda, Bm, ldb, C, ldc,
                bias, resid, K, alpha, relu ? 1 : 0, zInner,
                sAo, sAi, sBo, sBi, sCo, sCi);
    };
    auto ln = [&](const float* x, LNP p, float* out, int rows) {
        layernorm_kernel<<<dim3(rows), dim3(128), 0, stream>>>(x, p.a, p.b, out);
    };
    auto mha = [&](const AttnP& p, const float* Aq, const float* Akv,
                   const int* mask, int Sq, int Sk, float* X) {
        const int Tq = BATCH * Sq, Tk = BATCH * Sk;
        gemm(true, Aq,  DMODEL, p.q.w, DMODEL, Qb, DMODEL, Tq, DMODEL, DMODEL,
             p.q.b, 0, 1.f, false, 1, 1, 0,0,0,0,0,0);
        gemm(true, Akv, DMODEL, p.k.w, DMODEL, Kb, DMODEL, Tk, DMODEL, DMODEL,
             p.k.b, 0, 1.f, false, 1, 1, 0,0,0,0,0,0);
        gemm(true, Akv, DMODEL, p.v.w, DMODEL, Vb, DMODEL, Tk, DMODEL, DMODEL,
             p.v.b, 0, 1.f, false, 1, 1, 0,0,0,0,0,0);
        // scores[b,h] = (Q_bh @ K_bh^T) / sqrt(DK)    (batched over z = b*H+h)
        gemm(true, Qb, DMODEL, Kb, DMODEL, SC, Sk, Sq, Sk, DHEAD,
             0, 0, 0.125f, false, BATCH * NH, NH,
             (long)Sq * DMODEL, DHEAD, (long)Sk * DMODEL, DHEAD,
             (long)NH * Sq * Sk, (long)Sq * Sk);
        softmax_kernel<<<dim3(BATCH * NH * Sq), dim3(128), 0, stream>>>(SC, mask, Sq, Sk);
        // ctx[b,h] = P_bh @ V_bh
        gemm(false, SC, Sk, Vb, DMODEL, ATT, DMODEL, Sq, DHEAD, Sk,
             0, 0, 1.f, false, BATCH * NH, NH,
             (long)NH * Sq * Sk, (long)Sq * Sk, (long)Sk * DMODEL, DHEAD,
             (long)Sq * DMODEL, DHEAD);
        // X = X + ctx @ Wo^T + bo
        gemm(true, ATT, DMODEL, p.o.w, DMODEL, X, DMODEL, Tq, DMODEL, DMODEL,
             p.o.b, X, 1.f, false, 1, 1, 0,0,0,0,0,0);
    };
    auto ffn = [&](LinP l1, LinP l2, const float* nb, float* X, int T) {
        gemm(true, nb,  DMODEL, l1.w, DMODEL, FFH, DFFN, T, DFFN, DMODEL,
             l1.b, 0, 1.f, true, 1, 1, 0,0,0,0,0,0);
        gemm(true, FFH, DFFN, l2.w, DFFN, X, DMODEL, T, DMODEL, DFFN,
             l2.b, X, 1.f, false, 1, 1, 0,0,0,0,0,0);
    };

    // ---- embeddings ----
    const float scale = sqrtf((float)DMODEL);
    {
        long n = (long)Te * DMODEL;
        embed_kernel<<<dim3((unsigned)((n + 255) / 256)), dim3(256), 0, stream>>>(
            srcTok, src_emb, XE, (int)Te, SSRC, scale);
        n = (long)Td * DMODEL;
        embed_kernel<<<dim3((unsigned)((n + 255) / 256)), dim3(256), 0, stream>>>(
            tgtTok, tgt_emb, XD, (int)Td, STGT, scale);
    }

    // ---- encoder ----
    for (int l = 0; l < NL; ++l) {
        ln(XE, enc[l].sa_ln, NBF, (int)Te);
        mha(enc[l].sa, NBF, NBF, (const int*)0, SSRC, SSRC, XE);
        ln(XE, enc[l].ff_ln, NBF, (int)Te);
        ffn(enc[l].l1, enc[l].l2, NBF, XE, (int)Te);
    }
    ln(XE, enc_lnp, MEM, (int)Te);

    // ---- decoder ----
    for (int l = 0; l < NL; ++l) {
        ln(XD, dec[l].sa_ln, NBF, (int)Td);
        mha(dec[l].sa, NBF, NBF, decMask, STGT, STGT, XD);
        ln(XD, dec[l].ca_ln, NBF, (int)Td);
        mha(dec[l].ca, NBF, MEM, (const int*)0, STGT, SSRC, XD);
        ln(XD, dec[l].ff_ln, NBF, (int)Td);
        ffn(dec[l].l1, dec[l].l2, NBF, XD, (int)Td);
    }
    ln(XD, dec_lnp, NBF, (int)Td);

    // ---- generator + log-softmax ----
    float* out = (float*)d_out;
    gemm(true, NBF, DMODEL, genp.w, DMODEL, out, NVOCAB, (int)Td, NVOCAB, DMODEL,
         genp.b, 0, 1.f, false, 1, 1, 0,0,0,0,0,0);
    logsoftmax_kernel<<<dim3((int)Td), dim3(256), 0, stream>>>(out);
}